// Multi_adv_34840774705457
// MI455X (gfx1250) — compile-verified
//
#include <hip/hip_runtime.h>
#include <math.h>

typedef float v2f __attribute__((ext_vector_type(2)));
typedef float v8f __attribute__((ext_vector_type(8)));

#define H_DIM 192
#define BN_EPSF 1e-5f

// ---------------------------------------------------------------------------
// GEMM + fused column stats:
//   out[r,:] = act(in[r,:]) @ W + bias, act = optional (x*scale+shift, ReLU)
//   statP[blk, 0:192]   = sum of out over this block's rows (per column)
//   statP[blk, 192:384] = sum of out^2 over this block's rows (per column)
// Block: 128 threads = 4 waves. Tile: 64 rows x 192 cols. K = 192 in steps of 4
// via V_WMMA_F32_16X16X4_F32. Safe with out == A (in-place): all global reads
// of this block's rows happen before the barrier; stores are after.
// Deterministic: fixed-order register sums + shuffle + fixed-order LDS sum.
// ---------------------------------------------------------------------------
__global__ __launch_bounds__(128)
void gemm_bn_kernel(const float* __restrict__ A, const float* __restrict__ W,
                    const float* __restrict__ bias,
                    const float* __restrict__ scale, const float* __restrict__ shift,
                    int doBN, int N, float* __restrict__ out,
                    float* __restrict__ statP)
{
    __shared__ float sX[64][196];          // 196 stride: conflict-free column reads
    __shared__ float sSum[4][H_DIM];
    __shared__ float sSq[4][H_DIM];
    const int tid = threadIdx.x;
    const int rowBase = blockIdx.x * 64;

    for (int idx = tid; idx < 64 * H_DIM; idx += 128) {
        const int r = idx / H_DIM;
        const int c = idx - r * H_DIM;
        float v = 0.0f;
        if (rowBase + r < N) {
            v = A[(size_t)(rowBase + r) * H_DIM + c];
            if (doBN) v = fmaxf(0.0f, v * scale[c] + shift[c]);
        }
        sX[r][c] = v;
    }
    __syncthreads();

    const int lane = tid & 31;
    const int wave = tid >> 5;
    const int ml   = lane & 15;   // A: M within strip; B/C: N within 16-col tile
    const int kh   = lane >> 4;   // K-pair selector (A/B), +8-row selector (C)

    v8f acc[12];
#pragma unroll
    for (int t = 0; t < 12; ++t) acc[t] = (v8f)0.0f;

    const int arow = wave * 16 + ml;
    for (int k0 = 0; k0 < H_DIM; k0 += 4) {
        const int k = k0 + 2 * kh;
        v2f a;
        a.x = sX[arow][k];
        a.y = sX[arow][k + 1];
#pragma unroll
        for (int t = 0; t < 12; ++t) {
            v2f b;
            b.x = W[(size_t)k * H_DIM + 16 * t + ml];
            b.y = W[(size_t)(k + 1) * H_DIM + 16 * t + ml];
            acc[t] = __builtin_amdgcn_wmma_f32_16x16x4_f32(
                false, a, false, b, (short)0, acc[t], false, false);
        }
    }

    // Epilogue: bias add, store, and fused per-column partial stats.
#pragma unroll
    for (int t = 0; t < 12; ++t) {
        const int col = 16 * t + ml;
        const float bv = bias[col];
        float cs = 0.0f, cq = 0.0f;
#pragma unroll
        for (int v = 0; v < 8; ++v) {
            const int r = rowBase + wave * 16 + v + 8 * kh;
            const float z = acc[t][v] + bv;
            if (r < N) {
                out[(size_t)r * H_DIM + col] = z;
                cs += z;
                cq += z * z;
            }
        }
        // merge lane L with L+16 (same column, other 8 rows)
        cs += __shfl_down(cs, 16, 32);
        cq += __shfl_down(cq, 16, 32);
        if (lane < 16) {
            sSum[wave][col] = cs;
            sSq[wave][col]  = cq;
        }
    }
    __syncthreads();

    for (int c = tid; c < H_DIM; c += 128) {
        float ts = 0.0f, tq = 0.0f;
#pragma unroll
        for (int w = 0; w < 4; ++w) { ts += sSum[w][c]; tq += sSq[w][c]; }
        statP[(size_t)blockIdx.x * (2 * H_DIM) + c]         = ts;
        statP[(size_t)blockIdx.x * (2 * H_DIM) + H_DIM + c] = tq;
    }
}

// ---------------------------------------------------------------------------
// Reduce stat partials (fixed order) -> scale = g/sqrt(var+eps), shift = be-mean*scale
// ---------------------------------------------------------------------------
__global__ __launch_bounds__(192)
void bnparams_kernel(const float* __restrict__ partials, int nBlocks, int N,
                     const float* __restrict__ g, const float* __restrict__ be,
                     float* __restrict__ scale, float* __restrict__ shift)
{
    const int col = threadIdx.x;
    float s = 0.0f, q = 0.0f;
    for (int b = 0; b < nBlocks; ++b) {
        s += partials[(size_t)b * (2 * H_DIM) + col];
        q += partials[(size_t)b * (2 * H_DIM) + H_DIM + col];
    }
    const float invN = 1.0f / (float)N;
    const float mean = s * invN;
    const float var  = q * invN - mean * mean;
    const float sc   = g[col] / sqrtf(var + BN_EPSF);
    scale[col] = sc;
    shift[col] = be[col] - mean * sc;
}

// ---------------------------------------------------------------------------
// Final layer: BN2+ReLU -> @W3(192x8)+b3 -> sigmoid -> log_softmax -> NLL.
// One wave per row; lanes split 192 cols (6 each, coalesced). Deterministic
// shuffle-tree reductions; per-block partial sums of (lse - pred[label]).
// ---------------------------------------------------------------------------
__global__ __launch_bounds__(256)
void loss_kernel(const float* __restrict__ Z, const float* __restrict__ scale,
                 const float* __restrict__ shift, const float* __restrict__ W3,
                 const float* __restrict__ b3, const int* __restrict__ labels,
                 int N, float* __restrict__ partials)
{
    const int lane  = threadIdx.x & 31;
    const int wave  = threadIdx.x >> 5;
    const int gwave = blockIdx.x * 8 + wave;
    const int nwave = gridDim.x * 8;
    const int c0 = lane * 6;

    float w3r[6][8], sc[6], sh[6], b3r[8];
#pragma unroll
    for (int j = 0; j < 6; ++j) {
        sc[j] = scale[c0 + j];
        sh[j] = shift[c0 + j];
#pragma unroll
        for (int o = 0; o < 8; ++o) w3r[j][o] = W3[(size_t)(c0 + j) * 8 + o];
    }
#pragma unroll
    for (int o = 0; o < 8; ++o) b3r[o] = b3[o];

    float acc = 0.0f;
    for (int row = gwave; row < N; row += nwave) {
        float lg[8] = {0, 0, 0, 0, 0, 0, 0, 0};
#pragma unroll
        for (int j = 0; j < 6; ++j) {
            float h = Z[(size_t)row * H_DIM + c0 + j];
            h = fmaxf(0.0f, h * sc[j] + sh[j]);
#pragma unroll
            for (int o = 0; o < 8; ++o) lg[o] += h * w3r[j][o];
        }
#pragma unroll
        for (int o = 0; o < 8; ++o)
            for (int off = 16; off > 0; off >>= 1)
                lg[o] += __shfl_down(lg[o], off, 32);
        if (lane == 0) {
            float p[8];
            float m = -3.0e38f;
#pragma unroll
            for (int o = 0; o < 8; ++o) {
                p[o] = 1.0f / (1.0f + expf(-(lg[o] + b3r[o])));  // sigmoid
                m = fmaxf(m, p[o]);
            }
            float se = 0.0f;
#pragma unroll
            for (int o = 0; o < 8; ++o) se += expf(p[o] - m);
            const float lse = m + logf(se);
            acc += lse - p[labels[row]];   // -(log_softmax at label)
        }
    }

    __shared__ float wsum[8];
    if (lane == 0) wsum[wave] = acc;
    __syncthreads();
    if (threadIdx.x == 0) {
        float s = 0.0f;
        for (int w = 0; w < 8; ++w) s += wsum[w];
        partials[blockIdx.x] = s;
    }
}

__global__ __launch_bounds__(256)
void reduce_kernel(const float* __restrict__ partials, int n, float invN,
                   float* __restrict__ out)
{
    __shared__ float s[256];
    float v = 0.0f;
    for (int i = threadIdx.x; i < n; i += 256) v += partials[i];
    s[threadIdx.x] = v;
    __syncthreads();
    for (int off = 128; off > 0; off >>= 1) {
        if (threadIdx.x < off) s[threadIdx.x] += s[threadIdx.x + off];
        __syncthreads();
    }
    if (threadIdx.x == 0) out[0] = s[0] * invN;
}

// ---------------------------------------------------------------------------
extern "C" void kernel_launch(void* const* d_in, const int* in_sizes, int n_in,
                              void* d_out, int out_size, void* d_ws, size_t ws_size,
                              hipStream_t stream)
{
    const float* X   = (const float*)d_in[0];
    const int*   lab = (const int*)d_in[1];
    const float* W1  = (const float*)d_in[2];
    const float* b1  = (const float*)d_in[3];
    const float* g1  = (const float*)d_in[4];
    const float* be1 = (const float*)d_in[5];
    const float* W2  = (const float*)d_in[6];
    const float* b2  = (const float*)d_in[7];
    const float* g2  = (const float*)d_in[8];
    const float* be2 = (const float*)d_in[9];
    const float* W3  = (const float*)d_in[10];
    const float* b3  = (const float*)d_in[11];

    const int N = in_sizes[0] / H_DIM;                    // 400000
    const int gemmBlocks = (N + 63) / 64;                 // 6250
    const int lossBlocks = 256;

    // Workspace carve (256B-aligned chunks)
    char* w = (char*)d_ws;
    float* Z = (float*)w;       w += ((size_t)N * H_DIM * sizeof(float) + 255) & ~(size_t)255;
    float* statP = (float*)w;   w += ((size_t)gemmBlocks * 2 * H_DIM * sizeof(float) + 255) & ~(size_t)255;
    float* scale1 = (float*)w;  w += 256;
    float* shift1 = (float*)w;  w += 256;
    float* scale2 = (float*)w;  w += 256;
    float* shift2 = (float*)w;  w += 256;
    float* lossP = (float*)w;   w += ((size_t)lossBlocks * sizeof(float) + 255) & ~(size_t)255;

    // Layer 1: Z = X @ W1 + b1 (stats fused)
    gemm_bn_kernel<<<gemmBlocks, 128, 0, stream>>>(X, W1, b1, nullptr, nullptr, 0, N, Z, statP);
    bnparams_kernel<<<1, H_DIM, 0, stream>>>(statP, gemmBlocks, N, g1, be1, scale1, shift1);

    // Layer 2 (in-place): Z = relu(bn1(Z)) @ W2 + b2 (stats fused)
    gemm_bn_kernel<<<gemmBlocks, 128, 0, stream>>>(Z, W2, b2, scale1, shift1, 1, N, Z, statP);
    bnparams_kernel<<<1, H_DIM, 0, stream>>>(statP, gemmBlocks, N, g2, be2, scale2, shift2);

    // Layer 3 + loss
    loss_kernel<<<lossBlocks, 256, 0, stream>>>(Z, scale2, shift2, W3, b3, lab, N, lossP);
    reduce_kernel<<<1, 256, 0, stream>>>(lossP, lossBlocks, 1.0f / (float)N, (float*)d_out);
}